// MaskLinearAttention_61856118997252
// MI455X (gfx1250) — compile-verified
//
#include <hip/hip_runtime.h>
#include <hip/hip_bf16.h>
#include <math.h>
#include <stdint.h>

// Problem constants from the reference
#define BB 8
#define LL 4800
#define HH 8
#define DD 64
#define LC 32   // l-chunk for attn gemm
#define LB 64   // l-block for out gemm

typedef __attribute__((ext_vector_type(2))) float v2f;
typedef __attribute__((ext_vector_type(8))) float v8f;

// ---------------------------------------------------------------------------
// Kernel 1: attn[bh][d][e] = (sum_l q[b,l,h,d]*k[b,l,h,e]) / (||q_d|| ||k_e||)
// One workgroup per (b,h); 4 waves; WMMA f32 16x16x4 over K = L = 4800.
// Double-buffered async global->LDS staging (ASYNCcnt pipelined).
// ---------------------------------------------------------------------------
__global__ __launch_bounds__(128) void attn_gemm_kernel(const float* __restrict__ q,
                                                        const float* __restrict__ k,
                                                        float* __restrict__ attn)
{
    const int bh = blockIdx.x;
    const int b = bh / HH, h = bh % HH;
    __shared__ float Qc[2][LC * 64];
    __shared__ float Kc[2][LC * 64];
    __shared__ float red[128];
    __shared__ float rq[64];
    __shared__ float rk[64];

    const int t = threadIdx.x;
    const int wid = t >> 5, lane = t & 31;
    const int m0 = wid * 16;
    const int hi2 = (lane >> 4) * 2;   // 0 or 2 (K sub-offset)
    const int ml  = lane & 15;

    const float* qb = q + ((size_t)b * LL * HH + h) * 64;
    const float* kb = k + ((size_t)b * LL * HH + h) * 64;

    v8f acc[4] = {};
    float ssq = 0.0f;  // t<64: sumsq of q row t ; t>=64: sumsq of k row t-64

    const int NCH = LL / LC;  // 150 chunks

    // Issue async global->LDS loads for chunk c into buffer buf.
    // Per thread: 4 x b128 for Q + 4 x b128 for K  => 8 async instrs per wave.
    auto issue = [&](int c, int buf) {
        const float* qsrc = qb + (size_t)c * LC * HH * 64;
        const float* ksrc = kb + (size_t)c * LC * HH * 64;
        #pragma unroll
        for (int j = 0; j < 4; ++j) {
            const int idx = t + 128 * j;              // 0..511 (16B units)
            const int l = idx >> 4, c4 = idx & 15;
            const uint32_t goff = (uint32_t)((l * HH * 64 + c4 * 4) * sizeof(float));
            const uint32_t qlds = (uint32_t)(uintptr_t)&Qc[buf][l * 64 + c4 * 4];
            const uint32_t klds = (uint32_t)(uintptr_t)&Kc[buf][l * 64 + c4 * 4];
            asm volatile("global_load_async_to_lds_b128 %0, %1, %2"
                         :: "v"(qlds), "v"(goff), "s"(qsrc) : "memory");
            asm volatile("global_load_async_to_lds_b128 %0, %1, %2"
                         :: "v"(klds), "v"(goff), "s"(ksrc) : "memory");
        }
    };

    issue(0, 0);

    for (int c = 0; c < NCH; ++c) {
        const int buf = c & 1;
        if (c + 1 < NCH) {
            issue(c + 1, buf ^ 1);                       // prefetch next chunk
            asm volatile("s_wait_asynccnt 0x8" ::: "memory");  // chunk c done, c+1 in flight
        } else {
            asm volatile("s_wait_asynccnt 0x0" ::: "memory");
        }
        __syncthreads();

        // ---- fused sum-of-squares (for the L2 normalization) ----
        {
            const int d = t & 63;
            const float* base = (t < 64) ? Qc[buf] : Kc[buf];
            #pragma unroll 8
            for (int l = 0; l < LC; ++l) {
                const float x = base[l * 64 + d];
                ssq = fmaf(x, x, ssq);
            }
        }

        // ---- WMMA: each wave owns M strip m0..m0+15, all 4 N tiles ----
        #pragma unroll
        for (int ks = 0; ks < LC; ks += 4) {
            v2f a;
            a.x = Qc[buf][(ks + hi2    ) * 64 + m0 + ml];
            a.y = Qc[buf][(ks + hi2 + 1) * 64 + m0 + ml];
            #pragma unroll
            for (int nt = 0; nt < 4; ++nt) {
                v2f bf;
                bf.x = Kc[buf][(ks + hi2    ) * 64 + nt * 16 + ml];
                bf.y = Kc[buf][(ks + hi2 + 1) * 64 + nt * 16 + ml];
                acc[nt] = __builtin_amdgcn_wmma_f32_16x16x4_f32(
                    false, a, false, bf, (short)0, acc[nt], false, false);
            }
        }
        __syncthreads();   // reads of buf done before it is overwritten next iter
    }

    // ---- inverse norms ----
    red[t] = ssq;
    __syncthreads();
    if (t < 64) rq[t]      = 1.0f / fmaxf(sqrtf(red[t]), 1e-12f);
    else        rk[t - 64] = 1.0f / fmaxf(sqrtf(red[t]), 1e-12f);
    __syncthreads();

    // ---- scaled store: C/D layout -> M = m0 + j + 8*(lane>=16), N = nt*16 + (lane&15)
    float* out = attn + (size_t)bh * 64 * 64;
    const int hi = lane >> 4;
    #pragma unroll
    for (int nt = 0; nt < 4; ++nt) {
        #pragma unroll
        for (int j = 0; j < 8; ++j) {
            const int M = m0 + j + 8 * hi;
            const int N = nt * 16 + ml;
            out[M * 64 + N] = acc[nt][j] * rq[M] * rk[N];
        }
    }
}

// ---------------------------------------------------------------------------
// Kernel 2: gating + per-row top-k masked softmaxes collapsed into Acomb.
// One workgroup per (b,h); 64 threads; thread t owns row t.
// ---------------------------------------------------------------------------
__global__ __launch_bounds__(64) void gate_topk_kernel(const float* __restrict__ attn,
                                                       float* __restrict__ acomb,
                                                       const float* __restrict__ temperature,
                                                       const float* __restrict__ attns,
                                                       const float* __restrict__ row_w,
                                                       const float* __restrict__ row_b,
                                                       const float* __restrict__ col_w,
                                                       const float* __restrict__ col_b)
{
    const int bh = blockIdx.x;
    const int h = bh % HH;
    __shared__ float As[64][65];
    __shared__ float wcol[64];
    const int t = threadIdx.x;
    const float* a_in = attn + (size_t)bh * 4096;

    for (int i = t; i < 4096; i += 64) As[i >> 6][i & 63] = a_in[i];
    __syncthreads();

    // column gate (thread t = column e = t)
    float sc = 0.0f;
    for (int c = 0; c < 64; ++c) sc = fmaf(As[c][t], col_w[c], sc);
    wcol[t] = 1.0f / (1.0f + expf(-(sc + col_b[0])));
    // row gate for own row
    float sr = 0.0f;
    for (int e = 0; e < 64; ++e) sr = fmaf(As[t][e], row_w[e], sr);
    const float wrow = 1.0f / (1.0f + expf(-(sr + row_b[0])));
    __syncthreads();

    const float temp = temperature[h];
    float rowv[64];
    float mx = -INFINITY;
    for (int e = 0; e < 64; ++e) {
        const float v = As[t][e] * (wcol[e] + wrow) * temp;
        rowv[e] = v;
        mx = fmaxf(mx, v);
    }

    // per-thread descending insertion sort -> exact k-th largest thresholds
    float srt[64];
    for (int e = 0; e < 64; ++e) {
        const float v = rowv[e];
        int j = e;
        while (j > 0 && srt[j - 1] < v) { srt[j] = srt[j - 1]; --j; }
        srt[j] = v;
    }
    const float thr0 = srt[31];  // top 32  (C/2)
    const float thr1 = srt[41];  // top 42  (2C/3)
    const float thr2 = srt[47];  // top 48  (3C/4)
    const float thr3 = srt[50];  // top 51  (4C/5)

    float s0 = 0, s1 = 0, s2 = 0, s3 = 0;
    for (int e = 0; e < 64; ++e) {
        const float v = rowv[e];
        const float ev = expf(v - mx);
        if (v >= thr0) s0 += ev;
        if (v >= thr1) s1 += ev;
        if (v >= thr2) s2 += ev;
        if (v >= thr3) s3 += ev;
    }
    const float g0 = attns[0] / s0, g1 = attns[1] / s1;
    const float g2 = attns[2] / s2, g3 = attns[3] / s3;

    float* outp = acomb + (size_t)bh * 4096 + t * 64;
    for (int e = 0; e < 64; ++e) {
        const float v = rowv[e];
        const float ev = expf(v - mx);
        float r = 0.0f;
        if (v >= thr0) r = fmaf(g0, ev, r);
        if (v >= thr1) r = fmaf(g1, ev, r);
        if (v >= thr2) r = fmaf(g2, ev, r);
        if (v >= thr3) r = fmaf(g3, ev, r);
        outp[e] = r;
    }
}

// ---------------------------------------------------------------------------
// Kernel 3: out[b,l,h,d] = sum_e Acomb[bh][d][e] * v[b,l,h,e]
// Computed as Out[l][d] = V[l][:] . Acomb[d][:]  via WMMA (M=l, N=d, K=e=64).
// Grid: (L/LB, B*H); 4 waves per block, each wave owns a 16-row l strip.
// ---------------------------------------------------------------------------
__global__ __launch_bounds__(128) void out_gemm_kernel(const float* __restrict__ v,
                                                       const float* __restrict__ acomb,
                                                       float* __restrict__ out)
{
    const int bh = blockIdx.y;
    const int b = bh / HH, h = bh % HH;
    const int l0 = blockIdx.x * LB;
    __shared__ float At[64 * 64];   // At[e][d] = Acomb[d][e]
    __shared__ float Vt[64 * LB];   // Vt[e][l]

    const int t = threadIdx.x;
    const int wid = t >> 5, lane = t & 31;
    const int m0 = wid * 16;
    const int hi2 = (lane >> 4) * 2;
    const int ml  = lane & 15;

    // stage Acomb transposed
    const float* ac = acomb + (size_t)bh * 4096;
    for (int i = t; i < 1024; i += 128) {
        const int d = i >> 4, e4 = (i & 15) * 4;
        const float4 r = ((const float4*)(ac + d * 64))[i & 15];
        At[(e4    ) * 64 + d] = r.x;
        At[(e4 + 1) * 64 + d] = r.y;
        At[(e4 + 2) * 64 + d] = r.z;
        At[(e4 + 3) * 64 + d] = r.w;
    }
    // stage V strip, e-major
    const float* vb = v + ((size_t)b * LL * HH + h) * 64;
    for (int i = t; i < LB * 16; i += 128) {
        const int l = i >> 4, e4 = (i & 15) * 4;
        const float4 r = ((const float4*)(vb + (size_t)(l0 + l) * HH * 64))[i & 15];
        Vt[(e4    ) * LB + l] = r.x;
        Vt[(e4 + 1) * LB + l] = r.y;
        Vt[(e4 + 2) * LB + l] = r.z;
        Vt[(e4 + 3) * LB + l] = r.w;
    }
    __syncthreads();

    v8f acc[4] = {};
    #pragma unroll
    for (int ks = 0; ks < 64; ks += 4) {
        v2f a;
        a.x = Vt[(ks + hi2    ) * LB + m0 + ml];
        a.y = Vt[(ks + hi2 + 1) * LB + m0 + ml];
        #pragma unroll
        for (int nt = 0; nt < 4; ++nt) {
            v2f bf;
            bf.x = At[(ks + hi2    ) * 64 + nt * 16 + ml];
            bf.y = At[(ks + hi2 + 1) * 64 + nt * 16 + ml];
            acc[nt] = __builtin_amdgcn_wmma_f32_16x16x4_f32(
                false, a, false, bf, (short)0, acc[nt], false, false);
        }
    }

    // store: out[b, l0+M, h, N]; lanes 0-15 cover 64 contiguous bytes in d
    const int hi = lane >> 4;
    #pragma unroll
    for (int nt = 0; nt < 4; ++nt) {
        #pragma unroll
        for (int j = 0; j < 8; ++j) {
            const int M = m0 + j + 8 * hi;
            const int N = nt * 16 + ml;
            out[(((size_t)b * LL + l0 + M) * HH + h) * 64 + N] = acc[nt][j];
        }
    }
}

// ---------------------------------------------------------------------------
extern "C" void kernel_launch(void* const* d_in, const int* in_sizes, int n_in,
                              void* d_out, int out_size, void* d_ws, size_t ws_size,
                              hipStream_t stream)
{
    const float* q           = (const float*)d_in[0];
    const float* k           = (const float*)d_in[1];
    const float* v           = (const float*)d_in[2];
    const float* temperature = (const float*)d_in[3];
    const float* attns       = (const float*)d_in[4];
    const float* row_w       = (const float*)d_in[5];
    const float* row_b       = (const float*)d_in[6];
    const float* col_w       = (const float*)d_in[7];
    const float* col_b       = (const float*)d_in[8];
    float* out = (float*)d_out;

    float* attn  = (float*)d_ws;                              // 64*4096 floats = 1 MB
    float* acomb = (float*)d_ws + (size_t)BB * HH * 64 * 64;  // second 1 MB

    attn_gemm_kernel<<<BB * HH, 128, 0, stream>>>(q, k, attn);
    gate_topk_kernel<<<BB * HH, 64, 0, stream>>>(attn, acomb, temperature, attns,
                                                 row_w, row_b, col_w, col_b);
    out_gemm_kernel<<<dim3(LL / LB, BB * HH), 128, 0, stream>>>(v, acomb, out);
}